// StackedLSTMWithMultiAttn_79018808311848
// MI455X (gfx1250) — compile-verified
//
#include <hip/hip_runtime.h>
#include <hip/hip_bf16.h>
#include <math.h>

// Problem sizes (fixed by the reference)
#define NUM_LAYERS 2
#define RNN        512
#define BATCH      256
#define SRC        1024
#define FOUR_R     (4 * RNN)

#define NCHUNK     8                 // S-chunks per batch in attention pass 1
#define ROWS_PER_WG (SRC / NCHUNK)   // 128 rows per workgroup

typedef float v2f __attribute__((ext_vector_type(2)));
typedef float v4f __attribute__((ext_vector_type(4)));
typedef float v8f __attribute__((ext_vector_type(8)));

#define NEG_BIG (-1.0e30f)

__device__ __forceinline__ float sigmoidf_(float x) {
    return 1.0f / (1.0f + __expf(-x));
}

// ---------------------------------------------------------------------------
// WMMA f32 GEMM:  C[m][n] = sum_k A0[m][k]*W0[n][k] (+ A1[m][k]*W1[n][k])
//                          (+ bias0[n] + bias1[n])
// One wave (32 threads) computes one 16x16 tile. K in steps of 4 using
// V_WMMA_F32_16X16X4_F32.  A fragment: lanes 0-15 hold (M=lane, K=k..k+1),
// lanes 16-31 hold (M=lane-16, K=k+2..k+3) -> one float2 per lane.
// B fragment mirrors with N in lanes; B[k][n] = W[n][k], so each lane loads
// a float2 from row n of W at offset kk.
// ---------------------------------------------------------------------------
__global__ void gemm_wmma_f32(const float* __restrict__ A0,
                              const float* __restrict__ W0,
                              const float* __restrict__ A1,
                              const float* __restrict__ W1,
                              const float* __restrict__ bias0,
                              const float* __restrict__ bias1,
                              float* __restrict__ Cout,
                              int K, int ldc) {
    const int lane = threadIdx.x;          // 0..31 (wave32)
    const int half = lane >> 4;            // 0: K pair (k,k+1), 1: (k+2,k+3)
    const int l15  = lane & 15;
    const int n0   = blockIdx.x * 16;
    const int m0   = blockIdx.y * 16;

    const size_t arow = (size_t)(m0 + l15) * (size_t)K;
    const size_t brow = (size_t)(n0 + l15) * (size_t)K;

    v8f c = {};
    #pragma unroll 4
    for (int k = 0; k < K; k += 4) {
        const int kk = k + 2 * half;
        v2f a = *(const v2f*)(A0 + arow + kk);
        v2f b = *(const v2f*)(W0 + brow + kk);
        c = __builtin_amdgcn_wmma_f32_16x16x4_f32(false, a, false, b,
                                                  (short)0, c, false, false);
    }
    if (A1 != nullptr) {
        #pragma unroll 4
        for (int k = 0; k < K; k += 4) {
            const int kk = k + 2 * half;
            v2f a = *(const v2f*)(A1 + arow + kk);
            v2f b = *(const v2f*)(W1 + brow + kk);
            c = __builtin_amdgcn_wmma_f32_16x16x4_f32(false, a, false, b,
                                                      (short)0, c, false, false);
        }
    }

    float bv = 0.0f;
    if (bias0 != nullptr) bv += bias0[n0 + l15];
    if (bias1 != nullptr) bv += bias1[n0 + l15];

    // C/D layout: VGPR v -> (M = m0 + v + 8*half, N = n0 + l15)
    #pragma unroll
    for (int v = 0; v < 8; ++v) {
        const int m = m0 + v + 8 * half;
        Cout[(size_t)m * (size_t)ldc + n0 + l15] = c[v] + bv;
    }
}

// ---------------------------------------------------------------------------
// LSTM cell elementwise. gates layout (B, 4R) in i,f,g,o order.
// ---------------------------------------------------------------------------
__global__ void lstm_cell(const float* __restrict__ gates,
                          const float* __restrict__ c0,
                          float* __restrict__ c1_out,
                          float* __restrict__ h1_out) {
    const int idx = blockIdx.x * blockDim.x + threadIdx.x;   // B*R threads
    const int b = idx >> 9;          // /512
    const int r = idx & (RNN - 1);
    const float* g = gates + (size_t)b * FOUR_R;
    const float ig = sigmoidf_(g[r]);
    const float fg = sigmoidf_(g[r + RNN]);
    const float gg = tanhf(g[r + 2 * RNN]);
    const float og = sigmoidf_(g[r + 3 * RNN]);
    const float c  = fg * c0[idx] + ig * gg;
    const float h  = og * tanhf(c);
    c1_out[idx] = c;
    h1_out[idx] = h;
}

// ---------------------------------------------------------------------------
// Attention pass 1: one workgroup = (batch b, chunk of 128 source rows).
// 4 wave32s; each wave keeps t[b] resident (16 floats per lane) and streams
// ctx rows once: dot -> raw score (buffered), online softmax accumulation of
// the weighted context in registers. Wave partials merged in LDS; workgroup
// partial (m, l, acc[512]) written to scratch.
// ---------------------------------------------------------------------------
__global__ void attn_pass1(const float* __restrict__ ctx,   // (S, B, R)
                           const float* __restrict__ t,     // (B, R)
                           float* __restrict__ score_raw,   // (B, S) layer slice of d_out
                           float* __restrict__ pm,          // (B, NCHUNK)
                           float* __restrict__ pl,          // (B, NCHUNK)
                           float* __restrict__ pacc) {      // (B, NCHUNK, R)
    const int b     = blockIdx.y;
    const int chunk = blockIdx.x;
    const int lane  = threadIdx.x & 31;
    const int wave  = threadIdx.x >> 5;       // 0..3
    const int dbase = lane * 16;              // 16 d-values per lane

    __shared__ float sm[4];
    __shared__ float sl[4];
    __shared__ float sacc[4][RNN];

    float tr[16];
    #pragma unroll
    for (int j = 0; j < 16; ++j) tr[j] = t[(size_t)b * RNN + dbase + j];

    float acc[16];
    #pragma unroll
    for (int j = 0; j < 16; ++j) acc[j] = 0.0f;
    float m = NEG_BIG, l = 0.0f;

    for (int r = wave; r < ROWS_PER_WG; r += 4) {
        const int s = chunk * ROWS_PER_WG + r;
        const float* row = ctx + (size_t)s * BATCH * RNN + (size_t)b * RNN + dbase;
        float x[16];
        #pragma unroll
        for (int q = 0; q < 4; ++q) {
            v4f v = ((const v4f*)row)[q];
            x[4 * q + 0] = v.x; x[4 * q + 1] = v.y;
            x[4 * q + 2] = v.z; x[4 * q + 3] = v.w;
        }
        float dot = 0.0f;
        #pragma unroll
        for (int j = 0; j < 16; ++j) dot = fmaf(x[j], tr[j], dot);
        // butterfly reduce across wave32 -> result in all lanes
        #pragma unroll
        for (int off = 16; off > 0; off >>= 1)
            dot += __shfl_xor(dot, off, 32);

        if (lane == 0) score_raw[(size_t)b * SRC + s] = dot;

        // online softmax update
        const float mnew = fmaxf(m, dot);
        const float corr = __expf(m - mnew);     // 0 on first iteration
        const float w    = __expf(dot - mnew);
        l = l * corr + w;
        #pragma unroll
        for (int j = 0; j < 16; ++j) acc[j] = fmaf(acc[j], corr, w * x[j]);
        m = mnew;
    }

    // merge the 4 wave partials inside the workgroup
    #pragma unroll
    for (int j = 0; j < 16; ++j) sacc[wave][dbase + j] = acc[j];
    if (lane == 0) { sm[wave] = m; sl[wave] = l; }
    __syncthreads();

    const float M = fmaxf(fmaxf(sm[0], sm[1]), fmaxf(sm[2], sm[3]));
    float e[4];
    float L = 0.0f;
    #pragma unroll
    for (int w2 = 0; w2 < 4; ++w2) {
        e[w2] = __expf(sm[w2] - M);
        L += sl[w2] * e[w2];
    }
    const int d0 = threadIdx.x * 4;            // 128 threads * 4 = 512
    const size_t pbase = ((size_t)b * NCHUNK + chunk) * RNN;
    #pragma unroll
    for (int q = 0; q < 4; ++q) {
        const int d = d0 + q;
        pacc[pbase + d] = sacc[0][d] * e[0] + sacc[1][d] * e[1] +
                          sacc[2][d] * e[2] + sacc[3][d] * e[3];
    }
    if (threadIdx.x == 0) {
        pm[(size_t)b * NCHUNK + chunk] = M;
        pl[(size_t)b * NCHUNK + chunk] = L;
    }
}

// ---------------------------------------------------------------------------
// Attention combine: merge the NCHUNK chunk-partials per batch, divide by l*,
// apply residual h1 + wctx, write h output and next-layer x (and final x).
// ---------------------------------------------------------------------------
__global__ void attn_combine(const float* __restrict__ pm,
                             const float* __restrict__ pl,
                             const float* __restrict__ pacc,
                             const float* __restrict__ h1,
                             float* __restrict__ hout,
                             float* __restrict__ xbuf,
                             float* __restrict__ xout,      // may be null
                             float* __restrict__ mstar,
                             float* __restrict__ lstar) {
    const int b = blockIdx.x;
    float M = NEG_BIG;
    #pragma unroll
    for (int j = 0; j < NCHUNK; ++j) M = fmaxf(M, pm[(size_t)b * NCHUNK + j]);
    float e[NCHUNK];
    float L = 0.0f;
    #pragma unroll
    for (int j = 0; j < NCHUNK; ++j) {
        e[j] = __expf(pm[(size_t)b * NCHUNK + j] - M);
        L += pl[(size_t)b * NCHUNK + j] * e[j];
    }
    const float invL = 1.0f / L;

    const int d0 = threadIdx.x * 4;            // 128 threads * 4 = 512
    #pragma unroll
    for (int q = 0; q < 4; ++q) {
        const int d = d0 + q;
        float wc = 0.0f;
        #pragma unroll
        for (int j = 0; j < NCHUNK; ++j)
            wc += pacc[((size_t)b * NCHUNK + j) * RNN + d] * e[j];
        const float h = h1[(size_t)b * RNN + d] + wc * invL;
        hout[(size_t)b * RNN + d] = h;
        xbuf[(size_t)b * RNN + d] = h;
        if (xout != nullptr) xout[(size_t)b * RNN + d] = h;
    }
    if (threadIdx.x == 0) { mstar[b] = M; lstar[b] = L; }
}

// ---------------------------------------------------------------------------
// Normalize buffered raw scores into attention weights (in place in d_out).
// ---------------------------------------------------------------------------
__global__ void attn_norm(float* __restrict__ a,            // (B, S)
                          const float* __restrict__ mstar,
                          const float* __restrict__ lstar) {
    const int idx = blockIdx.x * blockDim.x + threadIdx.x;  // B*S threads
    const int b = idx >> 10;                                // /1024
    a[idx] = __expf(a[idx] - mstar[b]) / lstar[b];
}

// ---------------------------------------------------------------------------
extern "C" void kernel_launch(void* const* d_in, const int* in_sizes, int n_in,
                              void* d_out, int out_size, void* d_ws, size_t ws_size,
                              hipStream_t stream) {
    (void)in_sizes; (void)n_in; (void)out_size; (void)ws_size;

    const float* x_in    = (const float*)d_in[0];   // (B, R)
    const float* h0      = (const float*)d_in[1];   // (L, B, R)
    const float* c0      = (const float*)d_in[2];   // (L, B, R)
    const float* context = (const float*)d_in[3];   // (S, B, R)
    const float* W_ih    = (const float*)d_in[4];   // (L, 4R, R)
    const float* W_hh    = (const float*)d_in[5];   // (L, 4R, R)
    const float* b_ih    = (const float*)d_in[6];   // (L, 4R)
    const float* b_hh    = (const float*)d_in[7];   // (L, 4R)
    const float* W_attn  = (const float*)d_in[8];   // (L, R, R)

    float* out = (float*)d_out;
    const size_t BR = (size_t)BATCH * RNN;          // 131072
    const size_t BS = (size_t)BATCH * SRC;          // 262144
    float* x_out = out;                              // (B, R)
    float* h_out = out + BR;                         // (L, B, R)
    float* c_out = out + BR + NUM_LAYERS * BR;       // (L, B, R)
    float* a_out = out + BR + 2 * NUM_LAYERS * BR;   // (L, B, S)

    // workspace layout (floats)
    float* ws    = (float*)d_ws;
    float* gates = ws;                               // B*4R = 524288
    float* h1    = gates + (size_t)BATCH * FOUR_R;   // B*R
    float* tbuf  = h1 + BR;                          // B*R
    float* xbuf  = tbuf + BR;                        // B*R
    float* pm    = xbuf + BR;                        // B*NCHUNK
    float* pl    = pm + (size_t)BATCH * NCHUNK;      // B*NCHUNK
    float* mst   = pl + (size_t)BATCH * NCHUNK;      // B
    float* lst   = mst + BATCH;                      // B
    float* pacc  = lst + BATCH;                      // B*NCHUNK*R = 1048576

    const float* xcur = x_in;
    for (int i = 0; i < NUM_LAYERS; ++i) {
        const float* Wih_i = W_ih + (size_t)i * FOUR_R * RNN;
        const float* Whh_i = W_hh + (size_t)i * FOUR_R * RNN;
        const float* bih_i = b_ih + (size_t)i * FOUR_R;
        const float* bhh_i = b_hh + (size_t)i * FOUR_R;
        const float* Wat_i = W_attn + (size_t)i * RNN * RNN;
        const float* h0_i  = h0 + (size_t)i * BR;
        const float* c0_i  = c0 + (size_t)i * BR;
        float* hout_i = h_out + (size_t)i * BR;
        float* cout_i = c_out + (size_t)i * BR;
        float* aout_i = a_out + (size_t)i * BS;

        // gates = xcur @ W_ih^T + h0_i @ W_hh^T + b_ih + b_hh   (256 x 2048)
        gemm_wmma_f32<<<dim3(FOUR_R / 16, BATCH / 16), 32, 0, stream>>>(
            xcur, Wih_i, h0_i, Whh_i, bih_i, bhh_i, gates, RNN, FOUR_R);

        // LSTM cell elementwise -> c1 (output), h1 (scratch)
        lstm_cell<<<(int)(BR / 256), 256, 0, stream>>>(gates, c0_i, cout_i, h1);

        // t = h1 @ W_attn^T   (256 x 512)
        gemm_wmma_f32<<<dim3(RNN / 16, BATCH / 16), 32, 0, stream>>>(
            h1, Wat_i, nullptr, nullptr, nullptr, nullptr, tbuf, RNN, RNN);

        // single-pass online-softmax attention over context
        attn_pass1<<<dim3(NCHUNK, BATCH), 128, 0, stream>>>(
            context, tbuf, aout_i, pm, pl, pacc);

        attn_combine<<<BATCH, 128, 0, stream>>>(
            pm, pl, pacc, h1, hout_i, xbuf,
            (i == NUM_LAYERS - 1) ? x_out : nullptr, mst, lst);

        attn_norm<<<(int)(BS / 256), 256, 0, stream>>>(aout_i, mst, lst);

        xcur = xbuf;
    }
}